// TransformerBlockQuantum_65481071409112
// MI455X (gfx1250) — compile-verified
//
#include <hip/hip_runtime.h>

typedef __attribute__((ext_vector_type(2))) float v2f;
typedef __attribute__((ext_vector_type(8))) float v8f;

// D = A(16x4 f32) * B(4x16 f32) + C(16x16 f32)   -- CDNA5 V_WMMA_F32_16X16X4_F32
static __device__ __forceinline__ v8f wmma_f32(v2f a, v2f b, v8f c) {
  return __builtin_amdgcn_wmma_f32_16x16x4_f32(false, a, false, b, (short)0, c,
                                               false, false);
}

// Async global->LDS copy (GLOBAL_LOAD_ASYNC_TO_LDS_B32, ASYNCcnt-tracked).
// ldsoff = wave-relative LDS byte offset (low 32 bits of generic pointer),
// gaddr  = 64-bit global address.
static __device__ __forceinline__ void async_g2l_b32(unsigned ldsoff,
                                                     unsigned long long gaddr) {
  asm volatile("global_load_async_to_lds_b32 %0, %1, off"
               :
               : "v"(ldsoff), "v"(gaddr)
               : "memory");
}
static __device__ __forceinline__ void wait_asynccnt0() {
  asm volatile("s_wait_asynccnt 0" ::: "memory");
}

// ---- DPP16 butterfly reductions over 16-lane groups (pure VALU, no LDS) ----
// xor1 = quad_perm(1,0,3,2)=0xB1, xor2 = quad_perm(2,3,0,1)=0x4E,
// then row_half_mirror (0x141) and row_mirror (0x140) complete the 16-group.
template <int CTRL>
static __device__ __forceinline__ float dpp_f32(float v) {
  return __int_as_float(__builtin_amdgcn_update_dpp(
      0, __float_as_int(v), CTRL, 0xf, 0xf, true));
}
static __device__ __forceinline__ float redmax16(float v) {
  v = fmaxf(v, dpp_f32<0xB1>(v));
  v = fmaxf(v, dpp_f32<0x4E>(v));
  v = fmaxf(v, dpp_f32<0x141>(v));
  v = fmaxf(v, dpp_f32<0x140>(v));
  return v;
}
static __device__ __forceinline__ float redsum16(float v) {
  v += dpp_f32<0xB1>(v);
  v += dpp_f32<0x4E>(v);
  v += dpp_f32<0x141>(v);
  v += dpp_f32<0x140>(v);
  return v;
}

// ---------------------------------------------------------------------------
// Tiny GEMM:  out[M,N] = A[M,K] @ W[N,K]^T + bias  (optional ReLU).
// One wave per 16x16 tile; K,N compile-time so the loop is 3 straight WMMAs
// with branch-free loads (N-edge handled by index clamp: junk columns >= N
// are computed but never stored).
// ---------------------------------------------------------------------------
template <int K, int N, bool RELU>
__global__ __launch_bounds__(256) void gemm_bias_kernel(
    const float* __restrict__ A, const float* __restrict__ W,
    const float* __restrict__ bias, float* __restrict__ out, int M, int ntn) {
  int wave = threadIdx.x >> 5;
  int lane = threadIdx.x & 31;
  int half = lane >> 4;
  int lm = lane & 15;
  int tile = blockIdx.x * 8 + wave;
  int mt = tile / ntn, nt = tile % ntn;
  if (mt * 16 >= M) return;  // wave-uniform
  int m = mt * 16 + lm;      // A-frag row: M = lane&15
  int n = nt * 16 + lm;      // output col / W row for B-frag
  int nc = (n < N) ? n : (N - 1);  // clamped W row (junk cols never stored)

  v8f acc = {};
#pragma unroll
  for (int k0 = 0; k0 < K; k0 += 4) {
    int ka = k0 + 2 * half;  // frag K: elem0 = 2*half, elem1 = +1
    v2f a, b;
    a.x = (ka < K) ? A[m * K + ka] : 0.f;          // static: always true K=12
    a.y = (ka + 1 < K) ? A[m * K + ka + 1] : 0.f;  // static
    b.x = (ka < K) ? W[nc * K + ka] : 0.f;
    b.y = (ka + 1 < K) ? W[nc * K + ka + 1] : 0.f;
    acc = wmma_f32(a, b, acc);
  }
  if (n < N) {
    float bb = bias[n];
#pragma unroll
    for (int j = 0; j < 8; ++j) {  // C layout: M = j + 8*half, N = lane&15
      int row = mt * 16 + j + 8 * half;
      float v = acc[j] + bb;
      if (RELU) v = fmaxf(v, 0.f);
      out[row * N + n] = v;
    }
  }
}

// ---------------------------------------------------------------------------
// Flash attention, one (batch, head) per block. T=256, hd=3 (padded to 4).
// q/k/v staged into LDS with async global->LDS b32 copies; scores via WMMA
// 16x16x4; online softmax with DPP16 reductions (a C-tile row lives in one
// VGPR across 16 lanes); P relaid C->A layout via per-wave LDS bounce;
// P@V accumulated with 4 more WMMAs per tile.
// ---------------------------------------------------------------------------
__global__ __launch_bounds__(256) void attn_kernel(
    const float* __restrict__ qkv,  // [2048, 36] (q|k|v, 12 each)
    const int* __restrict__ mask,   // [8, 256]
    float* __restrict__ headout) {  // [2048, 12]
  __shared__ float qs[256][4];
  __shared__ float ks[256][4];
  __shared__ float vs[256][4];
  __shared__ float msk[256];
  __shared__ float pxch[8][16][16];

  int b = blockIdx.x >> 2;
  int h = blockIdx.x & 3;
  int tid = threadIdx.x;
  const float rsc = 0.57735026919f;  // 1/sqrt(3), folded into score scaling

  {
    int t = tid;
    const float* gbase = qkv + (b * 256 + t) * 36 + h * 3;
    // async DMA the 3 live floats of each of q,k,v into LDS rows
    async_g2l_b32((unsigned)(unsigned long long)&qs[t][0],
                  (unsigned long long)gbase);
    async_g2l_b32((unsigned)(unsigned long long)&qs[t][1],
                  (unsigned long long)(gbase + 1));
    async_g2l_b32((unsigned)(unsigned long long)&qs[t][2],
                  (unsigned long long)(gbase + 2));
    async_g2l_b32((unsigned)(unsigned long long)&ks[t][0],
                  (unsigned long long)(gbase + 12));
    async_g2l_b32((unsigned)(unsigned long long)&ks[t][1],
                  (unsigned long long)(gbase + 13));
    async_g2l_b32((unsigned)(unsigned long long)&ks[t][2],
                  (unsigned long long)(gbase + 14));
    async_g2l_b32((unsigned)(unsigned long long)&vs[t][0],
                  (unsigned long long)(gbase + 24));
    async_g2l_b32((unsigned)(unsigned long long)&vs[t][1],
                  (unsigned long long)(gbase + 25));
    async_g2l_b32((unsigned)(unsigned long long)&vs[t][2],
                  (unsigned long long)(gbase + 26));
    qs[t][3] = 0.f;  // K-dim zero pad
    ks[t][3] = 0.f;
    vs[t][3] = 0.f;
    msk[t] = (mask[b * 256 + t] != 0) ? -1e30f : 0.f;
  }
  wait_asynccnt0();
  __syncthreads();

  int wave = tid >> 5, lane = tid & 31, half = lane >> 4, lm = lane & 15;

  for (int rt = wave; rt < 16; rt += 8) {  // every wave: exactly 2 row tiles
    int m0 = rt * 16;
    v2f afrag;
    afrag.x = qs[m0 + lm][2 * half];
    afrag.y = qs[m0 + lm][2 * half + 1];

    v8f acc = {};
    float mx[8], lsum[8];
#pragma unroll
    for (int j = 0; j < 8; ++j) { mx[j] = -1e30f; lsum[j] = 0.f; }

    for (int ct = 0; ct < 16; ++ct) {
      int n0 = ct * 16;
      v2f bfrag;
      bfrag.x = ks[n0 + lm][2 * half];
      bfrag.y = ks[n0 + lm][2 * half + 1];
      v8f z = {};
      v8f s = wmma_f32(afrag, bfrag, z);  // 16x16 raw q.k tile
      float madd = msk[n0 + lm];          // column(key)-dependent
#pragma unroll
      for (int j = 0; j < 8; ++j) {  // row r = j + 8*half spans 16 lanes
        float sj = s[j] * rsc + madd;
        float rmax = redmax16(sj);
        float newm = fmaxf(mx[j], rmax);
        float scale = __expf(mx[j] - newm);
        float p = __expf(sj - newm);
        float rs = redsum16(p);
        lsum[j] = lsum[j] * scale + rs;
        acc[j] = acc[j] * scale;
        mx[j] = newm;
        s[j] = p;
      }
      __syncthreads();  // uniform across waves
#pragma unroll
      for (int j = 0; j < 8; ++j) pxch[wave][j + 8 * half][lm] = s[j];
      __syncthreads();
#pragma unroll
      for (int kk = 0; kk < 4; ++kk) {  // P(16x16) @ V(16x4pad), 4 K-steps
        int kbase = 4 * kk + 2 * half;
        v2f pa, pb;
        pa.x = pxch[wave][lm][kbase];
        pa.y = pxch[wave][lm][kbase + 1];
        // clamp col index: cols 3..15 produce junk, never stored
        pb.x = vs[n0 + kbase][lm & 3];
        pb.y = vs[n0 + kbase + 1][lm & 3];
        acc = wmma_f32(pa, pb, acc);
      }
    }
    if (lm < 3) {
#pragma unroll
      for (int j = 0; j < 8; ++j) {
        int t = m0 + j + 8 * half;
        float denom = (lsum[j] > 0.f) ? lsum[j] : 1.f;
        headout[(b * 256 + t) * 12 + h * 3 + lm] = acc[j] / denom;
      }
    }
  }
}

// ---------------------------------------------------------------------------
// 12-qubit state-vector circuit, one token per block, state (32KB) in LDS.
// CIRC==0: fused RX(x_w + theta_w). CIRC==1: fused RY(theta_w) @ RX(x_w).
// Then one CNOT ring (n_layers-1 == 1) and MeasureAll(PauliZ).
// ---------------------------------------------------------------------------
template <int CIRC>
__global__ __launch_bounds__(256) void qcirc_kernel(
    const float* __restrict__ ang,    // [2048, 12]
    const float* __restrict__ theta,  // [12]
    float* __restrict__ out) {        // [2048, 12]
  __shared__ float sr[4096];
  __shared__ float si[4096];
  __shared__ float G[12][8];  // per-wire 2x2 complex gate (re/im pairs)
  int tok = blockIdx.x;
  int tid = threadIdx.x;

  if (tid == 0) __builtin_prefetch(&ang[tok * 12], 0, 3);

  if (tid < 12) {
    int w = tid;
    float a = ang[tok * 12 + w];
    if (CIRC == 0) {  // RX(theta)*RX(a) = RX(a+theta)
      float ph = 0.5f * (a + theta[w]);
      float c = __cosf(ph), s = __sinf(ph);
      G[w][0] = c;   G[w][1] = 0.f;
      G[w][2] = 0.f; G[w][3] = -s;
      G[w][4] = 0.f; G[w][5] = -s;
      G[w][6] = c;   G[w][7] = 0.f;
    } else {  // RY(b) @ RX(a)
      float ca = __cosf(0.5f * a), sa = __sinf(0.5f * a);
      float tb = theta[w];
      float cb = __cosf(0.5f * tb), sb = __sinf(0.5f * tb);
      G[w][0] = cb * ca;  G[w][1] = sb * sa;
      G[w][2] = -sb * ca; G[w][3] = -cb * sa;
      G[w][4] = sb * ca;  G[w][5] = -cb * sa;
      G[w][6] = cb * ca;  G[w][7] = -sb * sa;
    }
  }
  for (int i = tid; i < 4096; i += 256) {
    sr[i] = (i == 0) ? 1.f : 0.f;
    si[i] = 0.f;
  }
  __syncthreads();

  // 12 fused single-qubit gates (wire w has amplitude stride 2^(11-w))
  for (int w = 0; w < 12; ++w) {
    int stride = 1 << (11 - w);
    float g00r = G[w][0], g00i = G[w][1], g01r = G[w][2], g01i = G[w][3];
    float g10r = G[w][4], g10i = G[w][5], g11r = G[w][6], g11i = G[w][7];
    for (int p = tid; p < 2048; p += 256) {
      int i0 = ((p & ~(stride - 1)) << 1) | (p & (stride - 1));
      int i1 = i0 + stride;
      float a0r = sr[i0], a0i = si[i0], a1r = sr[i1], a1i = si[i1];
      float n0r = g00r * a0r - g00i * a0i + g01r * a1r - g01i * a1i;
      float n0i = g00r * a0i + g00i * a0r + g01r * a1i + g01i * a1r;
      float n1r = g10r * a0r - g10i * a0i + g11r * a1r - g11i * a1i;
      float n1i = g10r * a0i + g10i * a0r + g11r * a1i + g11i * a1r;
      sr[i0] = n0r; si[i0] = n0i;
      sr[i1] = n1r; si[i1] = n1i;
    }
    __syncthreads();
  }

  // CNOT ring: (0,1)..(10,11),(11,0) — pure LDS swap where control==1
  for (int e = 0; e < 12; ++e) {
    int c = e, t = (e + 1) % 12;
    int pc = 11 - c, pt = 11 - t;
    int lo = (pc < pt) ? pc : pt;
    int hi = (pc < pt) ? pt : pc;
    for (int p = tid; p < 1024; p += 256) {
      int x = ((p >> lo) << (lo + 1)) | (p & ((1 << lo) - 1));
      x = ((x >> hi) << (hi + 1)) | (x & ((1 << hi) - 1));
      int i0 = x | (1 << pc);   // control=1, target=0
      int i1 = i0 | (1 << pt);  // control=1, target=1
      float tr = sr[i0]; sr[i0] = sr[i1]; sr[i1] = tr;
      float ti = si[i0]; si[i0] = si[i1]; si[i1] = ti;
    }
    __syncthreads();
  }

  // MeasureAll(PauliZ): out_w = sum_i sign_w(i) * |amp_i|^2
  float acc[12];
#pragma unroll
  for (int w = 0; w < 12; ++w) acc[w] = 0.f;
  for (int i = tid; i < 4096; i += 256) {
    float p = sr[i] * sr[i] + si[i] * si[i];
#pragma unroll
    for (int w = 0; w < 12; ++w) acc[w] += ((i >> (11 - w)) & 1) ? -p : p;
  }
  __syncthreads();
#pragma unroll
  for (int w = 0; w < 12; ++w) sr[tid * 12 + w] = acc[w];  // reuse state LDS
  __syncthreads();
  for (int s2 = 128; s2 > 0; s2 >>= 1) {
    if (tid < s2)
      for (int w = 0; w < 12; ++w)
        sr[tid * 12 + w] += sr[(tid + s2) * 12 + w];
    __syncthreads();
  }
  if (tid < 12) out[tok * 12 + tid] = sr[tid];
}

// ---------------------------------------------------------------------------
// LayerNorm over D=12: out = LN(base + add) * g + b ; one thread per token.
// ---------------------------------------------------------------------------
__global__ __launch_bounds__(256) void ln_kernel(
    const float* __restrict__ base, const float* __restrict__ add,
    const float* __restrict__ g, const float* __restrict__ bta,
    float* __restrict__ out, int ntok) {
  int n = blockIdx.x * blockDim.x + threadIdx.x;
  if (n >= ntok) return;
  float v[12];
  float m = 0.f;
#pragma unroll
  for (int i = 0; i < 12; ++i) {
    v[i] = base[n * 12 + i] + add[n * 12 + i];
    m += v[i];
  }
  m *= (1.f / 12.f);
  float var = 0.f;
#pragma unroll
  for (int i = 0; i < 12; ++i) {
    float d = v[i] - m;
    var += d * d;
  }
  var *= (1.f / 12.f);
  float r = rsqrtf(var + 1e-5f);
#pragma unroll
  for (int i = 0; i < 12; ++i)
    out[n * 12 + i] = (v[i] - m) * r * g[i] + bta[i];
}

// ---------------------------------------------------------------------------
extern "C" void kernel_launch(void* const* d_in, const int* in_sizes, int n_in,
                              void* d_out, int out_size, void* d_ws,
                              size_t ws_size, hipStream_t stream) {
  const float* x      = (const float*)d_in[0];   // [8,256,12]
  const float* Wi     = (const float*)d_in[1];   // [36,12]
  const float* bi     = (const float*)d_in[2];   // [36]
  const float* Wo     = (const float*)d_in[3];   // [12,12]
  const float* bo     = (const float*)d_in[4];   // [12]
  const float* atheta = (const float*)d_in[5];   // [12]
  const float* l1w    = (const float*)d_in[6];   // [12,12]
  const float* l1b    = (const float*)d_in[7];   // [12]
  const float* ftheta = (const float*)d_in[8];   // [12]
  const float* l2w    = (const float*)d_in[9];   // [12,12]
  const float* l2b    = (const float*)d_in[10];  // [12]
  const float* ln1g   = (const float*)d_in[11];
  const float* ln1b   = (const float*)d_in[12];
  const float* ln2g   = (const float*)d_in[13];
  const float* ln2b   = (const float*)d_in[14];
  const int*   mask   = (const int*)d_in[15];    // [8,256]

  float* ws = (float*)d_ws;
  float* qkv      = ws;               // 2048*36
  float* headout  = qkv + 73728;      // 2048*12
  float* attn_cls = headout + 24576;  // 2048*12
  float* attn_q   = attn_cls + 24576; // 2048*12
  float* x1       = attn_q + 24576;   // 2048*12
  float* q_in     = x1 + 24576;       // 2048*12
  float* ffn_q    = q_in + 24576;     // 2048*12
  float* ffn_lin  = ffn_q + 24576;    // 2048*12
  float* outp = (float*)d_out;        // 2048*12

  // 1) qkv = x @ Wi^T + bi   (M=2048, N=36, K=12; 384 tiles / 8 waves)
  gemm_bias_kernel<12, 36, false><<<48, 256, 0, stream>>>(x, Wi, bi, qkv, 2048,
                                                          3);
  // 2) multi-head flash attention (WMMA + async LDS staging)
  attn_kernel<<<32, 256, 0, stream>>>(qkv, mask, headout);
  // 3) attn_cls = headout @ Wo^T + bo
  gemm_bias_kernel<12, 12, false><<<16, 256, 0, stream>>>(headout, Wo, bo,
                                                          attn_cls, 2048, 1);
  // 4) quantum attention circuit (RX gates)
  qcirc_kernel<0><<<2048, 256, 0, stream>>>(attn_cls, atheta, attn_q);
  // 5) x1 = LN(x + attn_q)
  ln_kernel<<<8, 256, 0, stream>>>(x, attn_q, ln1g, ln1b, x1, 2048);
  // 6) q_in = x1 @ lin1_w^T + lin1_b
  gemm_bias_kernel<12, 12, false><<<16, 256, 0, stream>>>(x1, l1w, l1b, q_in,
                                                          2048, 1);
  // 7) quantum FFN circuit (RY∘RX gates)
  qcirc_kernel<1><<<2048, 256, 0, stream>>>(q_in, ftheta, ffn_q);
  // 8) ffn_lin = relu(ffn_q @ lin2_w^T + lin2_b)
  gemm_bias_kernel<12, 12, true><<<16, 256, 0, stream>>>(ffn_q, l2w, l2b,
                                                         ffn_lin, 2048, 1);
  // 9) out = LN(x1 + ffn_lin)
  ln_kernel<<<8, 256, 0, stream>>>(x1, ffn_lin, ln2g, ln2b, outp, 2048);
}